// VecKMSparseOptimized_58763742544835
// MI455X (gfx1250) — compile-verified
//
#include <hip/hip_runtime.h>
#include <hip/hip_bf16.h>

// ---------------- problem constants (from reference) ----------------
#define Hh 480
#define Ww 640
#define Dd 128
#define Kk 17
#define Rr 8
#define HP (Hh + 2 * Rr)            // 496 padded rows
#define WP (Ww + 2 * Rr)            // 656 padded cols
#define TD0u ((unsigned)(WP * Dd))  // tensor dim0 in 8-byte elements = 83968
#define TILE_ELEMS (Kk * Dd)        // 2176 elements per window row
#define CHUNK_ELEMS (TILE_ELEMS / 4)  // 544 elements per wave chunk
#define ROW_BYTES (TILE_ELEMS * 8)  // 17408 B per row buffer
#define CHUNK_BYTES (CHUNK_ELEMS * 8) // 4352 B per chunk

typedef float v2f __attribute__((ext_vector_type(2)));
typedef float v8f __attribute__((ext_vector_type(8)));
typedef unsigned int u32x4 __attribute__((ext_vector_type(4)));
typedef int i32x8 __attribute__((ext_vector_type(8)));
typedef int i32x4 __attribute__((ext_vector_type(4)));

#ifndef __has_builtin
#define __has_builtin(x) 0
#endif
#if __has_builtin(__builtin_amdgcn_wmma_f32_16x16x4_f32)
#define HAVE_WMMA_F32X4 1
#else
#define HAVE_WMMA_F32X4 0
#endif
#if __has_builtin(__builtin_amdgcn_tensor_load_to_lds) && \
    __has_builtin(__builtin_amdgcn_s_wait_tensorcnt)
#define HAVE_TDM 1
#else
#define HAVE_TDM 0
#endif

// ---------------- TDM helper: DMA one quarter of a window row into LDS ----
#if HAVE_TDM
__device__ __forceinline__ void tdm_load_chunk(const float* gbase, int row,
                                               int col0, int chunk,
                                               unsigned lds_byte) {
  unsigned long long ga =
      (unsigned long long)gbase +
      (((unsigned long long)row * (unsigned long long)WP +
        (unsigned long long)col0) * (unsigned long long)Dd +
       (unsigned long long)chunk * (unsigned long long)CHUNK_ELEMS) * 8ull;
  u32x4 g0;
  g0.x = 1u;                                        // count = 1 valid descriptor
  g0.y = lds_byte;                                  // LDS destination (bytes)
  g0.z = (unsigned)(ga & 0xffffffffull);            // global_addr[31:0] -> bits 95:64
  g0.w = (unsigned)((ga >> 32) & 0x1ffffffull)      // global_addr[56:32] -> bits 120:96
         | (2u << 30);                              // type = 2 ("image")
  i32x8 g1;
  g1[0] = (int)(3u << 16);                          // workgroup_mask=0, data_size=3 (8B)
  g1[1] = (int)((TD0u & 0xffffu) << 16);            // tensor_dim0[15:0] @ bits 63:48
  g1[2] = (int)((TD0u >> 16) | ((unsigned)HP << 16)); // dim0[31:16] | tensor_dim1 lo16
  g1[3] = (int)(((unsigned)CHUNK_ELEMS) << 16);     // tensor_dim1 hi16(0) | tile_dim0
  g1[4] = 1;                                        // tile_dim1=1, tile_dim2=0
  g1[5] = (int)TD0u;                                // tensor_dim0_stride[31:0]
  g1[6] = (int)((TD0u & 0xffffu) << 16);            // d0 stride hi16(0) | d1 stride lo16
  g1[7] = (int)(TD0u >> 16);                        // tensor_dim1_stride[47:16]
  i32x4 z4 = 0;
  i32x8 z8 = 0;
  // 6-arg form on this toolchain: (g0, g1, g2, g3, g4, cpol)
  __builtin_amdgcn_tensor_load_to_lds(g0, g1, z4, z4, z8, 0);
}
#endif

// ---------------- kernel weights: kw[k,d] = exp(i*(dx/R*X_d + dy/R*Y_d)) ---
__global__ __launch_bounds__(Dd) void kw_kernel(const float* __restrict__ X,
                                                const float* __restrict__ Y,
                                                float* __restrict__ kw) {
  const int k = blockIdx.x;   // 0..288, row-major (ky, kx)
  const int d = threadIdx.x;  // 0..127
  const int kyy = k / Kk - Rr;
  const int kxx = k % Kk - Rr;
  const float ph = ((float)kxx * (1.0f / (float)Rr)) * X[d] +
                   ((float)kyy * (1.0f / (float)Rr)) * Y[d];
  float s, c;
  __sincosf(ph, &s, &c);
  kw[((size_t)k * Dd + d) * 2 + 0] = c;
  kw[((size_t)k * Dd + d) * 2 + 1] = s;
}

// ---------------- phase-1: WMMA rank-1 phases + atomic scatter -------------
__global__ __launch_bounds__(32) void scatter_kernel(
    const float* __restrict__ t, const int* __restrict__ y,
    const int* __restrict__ x, const float* __restrict__ T,
    float* __restrict__ grid, float* __restrict__ counts, int N) {
  const int lane = threadIdx.x;                 // 0..31 (wave32)
  const long long base = (long long)blockIdx.x * 16;
  const int half = (lane >> 4) * 8;             // lanes 16-31 own rows M=8..15

  int cell[8];
  bool valid[8];
  for (int r = 0; r < 8; ++r) {
    long long m = base + half + r;
    valid[r] = (m < N);
    int yy = valid[r] ? y[m] : 0;
    int xx = valid[r] ? x[m] : 0;
    cell[r] = (yy + Rr) * WP + (xx + Rr);
  }
  if ((lane & 15) == 0) {
    for (int r = 0; r < 8; ++r)
      if (valid[r]) unsafeAtomicAdd(counts + cell[r], 1.0f);
  }

#if HAVE_WMMA_F32X4
  // A: 16x4 f32, only column K=0 populated with t values (rank-1 product)
  v2f A;
  {
    long long m = base + lane;
    A.x = (lane < 16 && m < N) ? t[m] : 0.0f;   // TEMPORAL_LENGTH == 1
    A.y = 0.0f;
  }
#else
  float tv[8];
  for (int r = 0; r < 8; ++r) {
    long long m = base + half + r;
    tv[r] = valid[r] ? t[m] : 0.0f;
  }
#endif

  for (int dt = 0; dt < 8; ++dt) {              // 8 tiles of 16 d-channels
    const int d = (lane & 15) + dt * 16;
#if HAVE_WMMA_F32X4
    v2f B;                                      // B: 4x16, row K=0 = T tile
    B.x = (lane < 16) ? T[dt * 16 + lane] : 0.0f;
    B.y = 0.0f;
    v8f C = {};
    C = __builtin_amdgcn_wmma_f32_16x16x4_f32(false, A, false, B, (short)0, C,
                                              false, false);
#else
    const float Td = T[d];
#endif
    for (int r = 0; r < 8; ++r) {
      if (!valid[r]) continue;
#if HAVE_WMMA_F32X4
      const float ph = C[r];                    // phase = t_m * T_d
#else
      const float ph = tv[r] * Td;
#endif
      float s, c;
      __sincosf(ph, &s, &c);
      float* p = grid + ((long long)cell[r] * Dd + d) * 2;
      unsafeAtomicAdd(p, c);                    // real += cos
      unsafeAtomicAdd(p + 1, s);                // imag += sin
    }
  }
}

// ---------------- phase-2: TDM-staged windowed complex reduce --------------
__global__ __launch_bounds__(Dd) void gather_kernel(
    const int* __restrict__ qy, const int* __restrict__ qx,
    const float* __restrict__ qt, const float* __restrict__ T,
    const float* __restrict__ grid, const float* __restrict__ counts,
    const float* __restrict__ kw, float* __restrict__ out) {
  __shared__ float lds_buf[2 * TILE_ELEMS * 2];  // double-buffered rows, 34816 B
  __shared__ float red[Dd];

  const int q = blockIdx.x;
  const int d = threadIdx.x;   // 0..127, one d-channel per thread
  const int wave = d >> 5;     // 4 waves, each DMAs one quarter row
  const int row0 = qy[q];      // padded window start row
  const int col0 = qx[q];      // padded window start col

  // ---- window count sum (289 cells, block tree-reduce) ----
  float cpart = 0.0f;
  for (int k = d; k < Kk * Kk; k += Dd) {
    const int kyy = k / Kk;
    const int kxx = k - kyy * Kk;
    cpart += counts[(size_t)(row0 + kyy) * WP + (col0 + kxx)];
  }
  red[d] = cpart;
  __syncthreads();
  for (int s = 64; s > 0; s >>= 1) {
    if (d < s) red[d] += red[d + s];
    __syncthreads();
  }
  const float cnt = red[0];

  float accr = 0.0f, acci = 0.0f;

#if HAVE_TDM
  const unsigned lds_base = (unsigned)(unsigned long long)(&lds_buf[0]);
  // prime pipeline: row 0 into buffer 0 (each wave its own quarter)
  tdm_load_chunk(grid, row0, col0, wave, lds_base + (unsigned)wave * CHUNK_BYTES);
  for (int ky = 0; ky < Kk; ++ky) {
    if (ky + 1 < Kk) {  // prefetch next row into other buffer, keep 1 in flight
      tdm_load_chunk(grid, row0 + ky + 1, col0, wave,
                     lds_base + (unsigned)((ky + 1) & 1) * ROW_BYTES +
                         (unsigned)wave * CHUNK_BYTES);
      __builtin_amdgcn_s_wait_tensorcnt(1);
    } else {
      __builtin_amdgcn_s_wait_tensorcnt(0);
    }
    __syncthreads();  // LDS row ky visible to all waves
    const float* buf = lds_buf + (ky & 1) * (TILE_ELEMS * 2);
    const float* kwr = kw + (size_t)(ky * Kk) * (Dd * 2);
    for (int kx = 0; kx < Kk; ++kx) {
      const float fr = buf[(kx * Dd + d) * 2 + 0];
      const float fi = buf[(kx * Dd + d) * 2 + 1];
      const float wr = kwr[(kx * Dd + d) * 2 + 0];
      const float wi = kwr[(kx * Dd + d) * 2 + 1];
      accr = fmaf(fr, wr, fmaf(-fi, wi, accr));
      acci = fmaf(fr, wi, fmaf(fi, wr, acci));
    }
    __syncthreads();  // done reading before this buffer is re-filled
  }
#else
  for (int ky = 0; ky < Kk; ++ky) {
    const float* rowp = grid + ((size_t)(row0 + ky) * WP + col0) * (Dd * 2);
    const float* kwr = kw + (size_t)(ky * Kk) * (Dd * 2);
    for (int kx = 0; kx < Kk; ++kx) {
      const float fr = rowp[(kx * Dd + d) * 2 + 0];
      const float fi = rowp[(kx * Dd + d) * 2 + 1];
      const float wr = kwr[(kx * Dd + d) * 2 + 0];
      const float wi = kwr[(kx * Dd + d) * 2 + 1];
      accr = fmaf(fr, wr, fmaf(-fi, wi, accr));
      acci = fmaf(fr, wi, fmaf(fi, wr, acci));
    }
  }
#endif

  // ---- re-center by query time, divide by count, store complex ----
  float s, c;
  __sincosf(-qt[q] * T[d], &s, &c);
  const float inv = 1.0f / fmaxf(cnt, 1.0f);
  const size_t o = ((size_t)q * Dd + d) * 2;
  out[o + 0] = (accr * c - acci * s) * inv;
  out[o + 1] = (accr * s + acci * c) * inv;
}

// ---------------- host-side launcher ---------------------------------------
extern "C" void kernel_launch(void* const* d_in, const int* in_sizes, int n_in,
                              void* d_out, int out_size, void* d_ws,
                              size_t ws_size, hipStream_t stream) {
  const float* t = (const float*)d_in[0];
  const int* y = (const int*)d_in[1];
  const int* x = (const int*)d_in[2];
  const int* qy = (const int*)d_in[3];
  const int* qx = (const int*)d_in[4];
  const float* qt = (const float*)d_in[5];
  const float* T = (const float*)d_in[6];
  const float* X = (const float*)d_in[7];
  const float* Y = (const float*)d_in[8];
  const int N = in_sizes[0];
  const int M = in_sizes[3];

  // workspace layout: [padded complex grid | padded counts | kw table]
  float* grid = (float*)d_ws;
  const size_t grid_f = (size_t)HP * WP * Dd * 2;  // 83,296,256 floats
  float* counts = grid + grid_f;
  const size_t cnt_f = (size_t)HP * WP;            // 325,376 floats
  float* kw = counts + cnt_f;                      // 289*128*2 floats

  (void)hipMemsetAsync(d_ws, 0, (grid_f + cnt_f) * sizeof(float), stream);
  kw_kernel<<<Kk * Kk, Dd, 0, stream>>>(X, Y, kw);
  scatter_kernel<<<(N + 15) / 16, 32, 0, stream>>>(t, y, x, T, grid, counts, N);
  gather_kernel<<<M, Dd, 0, stream>>>(qy, qx, qt, T, grid, counts, kw,
                                      (float*)d_out);
}